// MCGRU_30605936951455
// MI455X (gfx1250) — compile-verified
//
#include <hip/hip_runtime.h>

// Problem constants (from the reference)
#define NF   17
#define H8   8          // hidden size
#define G3H  24         // 3*H gates
#define TSZ  512
#define BSZ  256
#define SUMD 59         // sum(DIMS)
#define MAXD 13
#define HPAD 16         // padded LDS row stride for h tile (branch-free stores)

typedef float v2f __attribute__((ext_vector_type(2)));
typedef float v8f __attribute__((ext_vector_type(8)));

// Per-feature input slice offsets / widths (prefix sums of DIMS)
__device__ __constant__ int c_off[NF] = {0, 2, 10, 22, 35, 47, 48, 49, 50, 51, 52, 53, 54, 55, 56, 57, 58};
__device__ __constant__ int c_dim[NF] = {2, 8, 12, 13, 12, 1, 1, 1, 1, 1, 1, 1, 1, 1, 1, 1, 1};

__device__ __forceinline__ float fast_sigmoid(float x) {
    // single v_rcp_f32; saturates via exp -> {0, inf}
    return __builtin_amdgcn_rcpf(1.0f + __expf(-x));
}
__device__ __forceinline__ float fast_tanh(float x) {
    // tanh(x) = 1 - 2/(exp(2x)+1); branch-free, saturates to +/-1
    return 1.0f - 2.0f * __builtin_amdgcn_rcpf(1.0f + __expf(2.0f * x));
}

// One wave (32 lanes) handles (feature f, 16-batch tile). h state lives in LDS+regs
// across the whole 512-step scan; both GEMMs per step are V_WMMA_F32_16X16X4_F32.
__global__ __launch_bounds__(32) void mcgru_wmma_kernel(
    const float* __restrict__ x,     // (BS, TS, 59)
    const float* __restrict__ Wih,   // (NF, 24, 13)
    const float* __restrict__ Whh,   // (NF, 24, 8)
    const float* __restrict__ bih,   // (NF, 24)
    const float* __restrict__ bhh,   // (NF, 24)
    float* __restrict__ out)         // (BS, TS, NF, 8)
{
    __shared__ float h_lds[16 * HPAD]; // h tile, row stride padded to 16 (cols 8..15 junk)

    const int lane = threadIdx.x;    // 0..31 (wave32)
    const int half = lane >> 4;      // 0 | 1
    const int l15  = lane & 15;
    const int f    = blockIdx.y;
    const int b0   = blockIdx.x * 16;
    const int off  = c_off[f];
    const int d    = c_dim[f];

    // ---- B operands (weights) in WMMA B-layout, resident in VGPRs for the scan ----
    // B (4x16 KxN): VGPR v holds row K = 4c + v + 2*half, column N = l15 (+16 per tile)
    v2f bx[2][4];   // [N-tile][K-chunk] for W_ih  (K padded 13->16; zeros for k>=d)
    v2f bhw[2][2];  // [N-tile][K-chunk] for W_hh  (K = 8)
#pragma unroll
    for (int T = 0; T < 2; ++T) {
        const int g = T * 16 + l15;
#pragma unroll
        for (int c = 0; c < 4; ++c) {
#pragma unroll
            for (int v = 0; v < 2; ++v) {
                const int k = 4 * c + v + 2 * half;
                bx[T][c][v] = (g < G3H && k < d) ? Wih[(f * G3H + g) * MAXD + k] : 0.0f;
            }
        }
#pragma unroll
        for (int c = 0; c < 2; ++c) {
#pragma unroll
            for (int v = 0; v < 2; ++v) {
                const int k = 4 * c + v + 2 * half;
                bhw[T][c][v] = (g < G3H) ? Whh[(f * G3H + g) * H8 + k] : 0.0f;
            }
        }
    }

    // Clamped per-lane x indices: loads become unconditional; k>=d rows of B are 0,
    // so any finite value loaded there contributes nothing to the WMMA sum.
    int kc[4][2];
#pragma unroll
    for (int c = 0; c < 4; ++c)
#pragma unroll
        for (int v = 0; v < 2; ++v) {
            const int k = 4 * c + v + 2 * half;
            kc[c][v] = (k < d) ? k : (d - 1);
        }

    // Biases folded into WMMA C: tile0 (r,z) sums gx+gh -> b_ih+b_hh; tile1 keeps
    // gx_n (b_ih) and gh_n (b_hh) separate because n = tanh(gx_n + r*gh_n).
    const float bias0 = bih[f * G3H + l15] + bhh[f * G3H + l15];
    const float biasX = (l15 < H8) ? bih[f * G3H + 16 + l15] : 0.0f;
    const float biasH = (l15 < H8) ? bhh[f * G3H + 16 + l15] : 0.0f;

    // ---- init h = 0 ----
#pragma unroll
    for (int i = 0; i < 8; ++i) h_lds[lane * 8 + i] = 0.0f;
    float hprev[8];
#pragma unroll
    for (int i = 0; i < 8; ++i) hprev[i] = 0.0f;
    __syncthreads();

    // x cursor for this lane's M row (A-layout rows duplicate across lane halves)
    const float* xp = x + ((size_t)(b0 + l15) * TSZ) * SUMD + off;
    // coalesced output cursor: lane -> (m = lane/2, j-quad = lane&1), float4 per step
    float* op = out + ((size_t)(b0 + (lane >> 1)) * TSZ) * (NF * H8) + f * H8 + (lane & 1) * 4;

    for (int t = 0; t < TSZ; ++t) {
        __builtin_prefetch(xp + SUMD, 0, 0);  // speculative prefetch of next step's row

        // A operand: x tile, 16xK (K padded to 16); unconditional clamped loads
        v2f ax[4];
#pragma unroll
        for (int c = 0; c < 4; ++c) {
            ax[c][0] = xp[kc[c][0]];
            ax[c][1] = xp[kc[c][1]];
        }
        // A operand: h_{t-1} tile, 16x8, re-shaped from LDS (2x ds_load_2addr_b64)
        v2f ha[2];
#pragma unroll
        for (int c = 0; c < 2; ++c) {
            ha[c][0] = h_lds[l15 * HPAD + 4 * c + 2 * half + 0];
            ha[c][1] = h_lds[l15 * HPAD + 4 * c + 2 * half + 1];
        }

        v8f acc0 = {bias0, bias0, bias0, bias0, bias0, bias0, bias0, bias0}; // g 0..15 (r|z), gx+gh
        v8f accX = {biasX, biasX, biasX, biasX, biasX, biasX, biasX, biasX}; // g 16..23, gx part
        v8f accH = {biasH, biasH, biasH, biasH, biasH, biasH, biasH, biasH}; // g 16..23, gh part

        // Input projection: 4 K-chunks x 2 N-tiles
#pragma unroll
        for (int c = 0; c < 4; ++c) {
            acc0 = __builtin_amdgcn_wmma_f32_16x16x4_f32(false, ax[c], false, bx[0][c],
                                                         (short)0, acc0, false, false);
            accX = __builtin_amdgcn_wmma_f32_16x16x4_f32(false, ax[c], false, bx[1][c],
                                                         (short)0, accX, false, false);
        }
        // Recurrent projection: 2 K-chunks x 2 N-tiles
#pragma unroll
        for (int c = 0; c < 2; ++c) {
            acc0 = __builtin_amdgcn_wmma_f32_16x16x4_f32(false, ha[c], false, bhw[0][c],
                                                         (short)0, acc0, false, false);
            accH = __builtin_amdgcn_wmma_f32_16x16x4_f32(false, ha[c], false, bhw[1][c],
                                                         (short)0, accH, false, false);
        }

        __syncthreads();  // all ha reads done before h_lds is overwritten

        // Gate nonlinearities. C-layout: element (M = i + 8*half, N = l15).
        // r(m,j) at lane j, z(m,j) at lane j+8 -> shuffle-xor-8 within halves.
        // Lanes l15>=8 write junk into padded LDS columns that are never read
        // (their accX/accH are exactly 0 -> tanh(0)=0, no NaN).
#pragma unroll
        for (int i = 0; i < 8; ++i) {
            const float s  = fast_sigmoid(acc0[i]);
            const float z  = __shfl_xor(s, 8, 32);
            const float n  = fast_tanh(accX[i] + s * accH[i]);
            const float hn = (1.0f - z) * n + z * hprev[i];
            hprev[i] = hn;
            h_lds[(i + 8 * half) * HPAD + l15] = hn;   // unconditional (padded stride)
        }
        __syncthreads();

        // Coalesced output: 32 lanes x float4 = this step's full 16x8 h tile
        const float4 o = *reinterpret_cast<const float4*>(
            &h_lds[(lane >> 1) * HPAD + (lane & 1) * 4]);
        *reinterpret_cast<float4*>(op) = o;

        xp += SUMD;
        op += NF * H8;
    }
}

extern "C" void kernel_launch(void* const* d_in, const int* in_sizes, int n_in,
                              void* d_out, int out_size, void* d_ws, size_t ws_size,
                              hipStream_t stream) {
    (void)in_sizes; (void)n_in; (void)out_size; (void)d_ws; (void)ws_size;
    const float* x    = (const float*)d_in[0];
    const float* Wih  = (const float*)d_in[1];
    const float* Whh  = (const float*)d_in[2];
    const float* bih  = (const float*)d_in[3];
    const float* bhh  = (const float*)d_in[4];
    float* out = (float*)d_out;

    dim3 grid(BSZ / 16, NF);   // 16 batch tiles x 17 features = 272 waves
    mcgru_wmma_kernel<<<grid, 32, 0, stream>>>(x, Wih, Whh, bih, bhh, out);
}